// POIEncoder_1142461301191
// MI455X (gfx1250) — compile-verified
//
#include <hip/hip_runtime.h>
#include <hip/hip_bf16.h>
#include <cstdint>

typedef __attribute__((ext_vector_type(2))) float v2f;
typedef __attribute__((ext_vector_type(8))) float v8f;

#define N_NODES 50000
#define N_EDGES 800000
#define IN_CH   256
#define HID_CH  128

// ---------------------------------------------------------------------------
// Kernel 1: init deg to 1.0 (self-loop weight) and out accumulator to 0.
// ---------------------------------------------------------------------------
__global__ void k_init(float* __restrict__ deg, float* __restrict__ out) {
    int idx = blockIdx.x * blockDim.x + threadIdx.x;
    if (idx < N_NODES * HID_CH) out[idx] = 0.0f;
    if (idx < N_NODES)          deg[idx] = 1.0f;   // self-loop weight
}

// ---------------------------------------------------------------------------
// Kernel 2: degree accumulation at destination (col) with edge weights.
// ---------------------------------------------------------------------------
__global__ void k_degree(const long long* __restrict__ ei,
                         const float* __restrict__ ew,
                         float* __restrict__ deg) {
    int e = blockIdx.x * blockDim.x + threadIdx.x;
    if (e >= N_EDGES) return;
    long long c = ei[(size_t)N_EDGES + e];          // col = edge_index[1][e]
    atomicAdd(&deg[c], ew[e]);
}

// ---------------------------------------------------------------------------
// Kernel 3: deg -> deg^{-1/2} in place.
// ---------------------------------------------------------------------------
__global__ void k_rsqrt(float* __restrict__ deg) {
    int n = blockIdx.x * blockDim.x + threadIdx.x;
    if (n >= N_NODES) return;
    float d = deg[n];
    deg[n] = (d > 0.0f) ? __frsqrt_rn(d) : 0.0f;
}

// ---------------------------------------------------------------------------
// Kernel 4: h = x @ W with fp32 WMMA (V_WMMA_F32_16X16X4_F32).
// Block = 256 threads = 8 wave32s. Block computes a 16-row x 128-col tile.
// A panel (16 x 256 fp32 = 16 KB) staged in LDS, shared by all 8 waves.
// Each wave owns one 16-wide N tile and runs 64 K-steps of 16x16x4.
//
// VGPR layouts (ISA 7.12.2):
//   A 16x4 fp32: lanes 0-15 -> M=lane, {K+0,K+1}; lanes 16-31 -> M=lane-16, {K+2,K+3}
//   B  4x16 fp32 (mirrored): lanes 0-15 -> N=lane, {K+0,K+1}; lanes 16-31 -> {K+2,K+3}
//   C/D 16x16 fp32: vgpr r: lanes 0-15 = row r, lanes 16-31 = row r+8; col = lane%16
// ---------------------------------------------------------------------------
__global__ void __launch_bounds__(256)
k_gemm_wmma(const float* __restrict__ X,   // [N_NODES, IN_CH]
            const float* __restrict__ W,   // [IN_CH, HID_CH]
            float* __restrict__ H) {       // [N_NODES, HID_CH]
    __shared__ float As[16 * IN_CH];       // 16 KB

    const int m0 = blockIdx.x * 16;

    // Cooperative, fully coalesced load of the 16x256 A panel (float4).
    const float4* Xv  = reinterpret_cast<const float4*>(X + (size_t)m0 * IN_CH);
    float4*       Asv = reinterpret_cast<float4*>(As);
    for (int i = threadIdx.x; i < 16 * IN_CH / 4; i += 256) Asv[i] = Xv[i];
    __syncthreads();

    const int wave = threadIdx.x >> 5;     // 0..7  -> N tile
    const int lane = threadIdx.x & 31;
    const int hsel = lane >> 4;            // 0: K+{0,1}, 1: K+{2,3}
    const int l16  = lane & 15;
    const int n0   = wave * 16;

    v8f acc = {};
    for (int k = 0; k < IN_CH; k += 4) {
        // A fragment from LDS
        const float* ap = &As[l16 * IN_CH + k + hsel * 2];
        v2f a; a.x = ap[0]; a.y = ap[1];
        // B fragment from global (W stays L2/WGP$-resident: only 128 KB)
        const float* bp = &W[(size_t)(k + hsel * 2) * HID_CH + n0 + l16];
        v2f b; b.x = bp[0]; b.y = bp[HID_CH];
        acc = __builtin_amdgcn_wmma_f32_16x16x4_f32(
            /*neg_a=*/false, a, /*neg_b=*/false, b,
            /*c_mod=*/(short)0, acc, /*reuse_a=*/false, /*reuse_b=*/false);
    }

    // Store D tile: per-VGPR, 16 consecutive lanes write 64 contiguous bytes.
    float* hp = H + (size_t)m0 * HID_CH + n0;
    #pragma unroll
    for (int r = 0; r < 8; ++r) {
        int m = r + hsel * 8;
        hp[(size_t)m * HID_CH + l16] = acc[r];
    }
}

// ---------------------------------------------------------------------------
// Kernel 5: edge scatter. One wave32 per edge; each lane moves a float4 so a
// wave covers the full 512 B destination row. ~820 MB of traffic -> the
// dominant phase; this layout keeps both the gather and the atomics fully
// coalesced (global_atomic_add_f32, no return).
// ---------------------------------------------------------------------------
__global__ void __launch_bounds__(256)
k_scatter(const float* __restrict__ H,
          const long long* __restrict__ ei,
          const float* __restrict__ ew,
          const float* __restrict__ dis,
          float* __restrict__ out) {
    int e = blockIdx.x * 8 + (threadIdx.x >> 5);
    if (e >= N_EDGES) return;
    int lane = threadIdx.x & 31;

    long long r = ei[e];                      // row = edge_index[0][e]
    long long c = ei[(size_t)N_EDGES + e];    // col = edge_index[1][e]
    float norm = dis[r] * ew[e] * dis[c];

    const float4* hv = reinterpret_cast<const float4*>(H + (size_t)r * HID_CH);
    float4 v = hv[lane];

    float* op = out + (size_t)c * HID_CH + lane * 4;
    atomicAdd(op + 0, v.x * norm);
    atomicAdd(op + 1, v.y * norm);
    atomicAdd(op + 2, v.z * norm);
    atomicAdd(op + 3, v.w * norm);
}

// ---------------------------------------------------------------------------
// Kernel 6: finalize. Self-loop contribution folded in analytically
// (h[n] * dis[n]^2), plus bias, then PReLU.
// ---------------------------------------------------------------------------
__global__ void k_finalize(const float* __restrict__ H,
                           const float* __restrict__ dis,
                           const float* __restrict__ bias,
                           const float* __restrict__ alpha,
                           float* __restrict__ out) {
    int idx = blockIdx.x * blockDim.x + threadIdx.x;
    if (idx >= N_NODES * HID_CH) return;
    int n  = idx >> 7;          // / HID_CH
    int ch = idx & (HID_CH - 1);
    float d = dis[n];
    float v = out[idx] + H[idx] * d * d + bias[ch];
    out[idx] = (v >= 0.0f) ? v : alpha[ch] * v;
}

// ---------------------------------------------------------------------------
extern "C" void kernel_launch(void* const* d_in, const int* in_sizes, int n_in,
                              void* d_out, int out_size, void* d_ws, size_t ws_size,
                              hipStream_t stream) {
    const float*     x     = (const float*)d_in[0];      // [50000, 256]
    const long long* ei    = (const long long*)d_in[1];  // [2, 800000] int64
    const float*     ew    = (const float*)d_in[2];      // [800000]
    const float*     W     = (const float*)d_in[3];      // [256, 128]
    const float*     bias  = (const float*)d_in[4];      // [128]
    const float*     alpha = (const float*)d_in[5];      // [128]
    float*           out   = (float*)d_out;              // [50000, 128]

    // Workspace layout: [deg/dis: 64K floats][H: 50000*128 floats]
    float* deg = (float*)d_ws;
    float* H   = (float*)d_ws + 65536;

    const int threads = 256;
    const int total   = N_NODES * HID_CH;                // 6.4M

    k_init<<<(total + threads - 1) / threads, threads, 0, stream>>>(deg, out);
    k_degree<<<(N_EDGES + threads - 1) / threads, threads, 0, stream>>>(ei, ew, deg);
    k_rsqrt<<<(N_NODES + threads - 1) / threads, threads, 0, stream>>>(deg);
    k_gemm_wmma<<<N_NODES / 16, threads, 0, stream>>>(x, W, H);   // 50000/16 = 3125
    k_scatter<<<(N_EDGES + 7) / 8, threads, 0, stream>>>(H, ei, ew, deg, out);
    k_finalize<<<(total + threads - 1) / threads, threads, 0, stream>>>(H, deg, bias, alpha, out);
}